// Linear_61615600828584
// MI455X (gfx1250) — compile-verified
//
#include <hip/hip_runtime.h>
#include <hip/hip_bf16.h>

#define B_        64
#define IN_       512
#define OUT_      512
#define TPO_      512          // tables per output feature (== IN_)
#define OPB_      2            // output features handled per block -> 256 blocks
#define NTHREADS  256
#define LDS_STRIDE 65          // padded stride -> conflict-free transposed layout

typedef __attribute__((address_space(1))) int gint_t;   // global int
typedef __attribute__((address_space(3))) int lint_t;   // LDS int

// ---- CDNA5 async global -> LDS copy (ASYNCcnt-tracked) ----------------------
__device__ __forceinline__ void async_ld_b32(const float* gsrc, float* ldst) {
#if defined(__AMDGCN__) && __has_builtin(__builtin_amdgcn_global_load_async_to_lds_b32)
    __builtin_amdgcn_global_load_async_to_lds_b32(
        (gint_t*)(int*)gsrc, (lint_t*)(int*)ldst, /*imm_offset=*/0, /*cpol=*/0);
#else
    unsigned loff = (unsigned)(__UINTPTR_TYPE__)(lint_t*)(int*)ldst;
    asm volatile("global_load_async_to_lds_b32 %0, %1, off"
                 :: "v"(loff), "v"(gsrc) : "memory");
#endif
}

__device__ __forceinline__ void wait_async0() {
#if defined(__AMDGCN__) && __has_builtin(__builtin_amdgcn_s_wait_asynccnt)
    __builtin_amdgcn_s_wait_asynccnt(0);
#else
    asm volatile("s_wait_asynccnt 0x0" ::: "memory");
#endif
}

// -----------------------------------------------------------------------------
// One block: OPB_ output features. Lanes = batch rows, so the table index is
// wave-uniform: masks/weights stream through the scalar/constant path, the
// u/v coefficient transform stays on the SALU pipe, and the vector pipe does
// only 2 conflict-free ds_loads + 6 VALU per table.
// -----------------------------------------------------------------------------
__global__ __launch_bounds__(NTHREADS)
void lut_linear_kernel(const float* __restrict__ input,     // [B, IN]
                       const int*   __restrict__ input_mask,// [T*2]
                       const float* __restrict__ lut_w,     // [T, 4]
                       const float* __restrict__ bias,      // [OUT]
                       float*       __restrict__ out)       // [B, OUT]
{
    __shared__ float xT[IN_ * LDS_STRIDE];   // transposed input: xT[i*65 + b]
    __shared__ float red[4 * B_];            // cross-g reduction buffer

    const int tid = threadIdx.x;

    // ---- Stage input into LDS, transposed, via async global->LDS -----------
    // Uniform trip count (128) -> clean scalar-controlled loop. Global reads
    // walk the array contiguously (coalesced); the per-lane LDS destination
    // performs the transpose; stride-65 rows keep every access conflict-free.
    #pragma unroll 4
    for (int it = 0; it < (B_ * IN_) / NTHREADS; ++it) {
        const int e  = it * NTHREADS + tid;
        const int bb = e >> 9;               // e / IN_
        const int ii = e & (IN_ - 1);        // e % IN_
        async_ld_b32(input + e, &xT[ii * LDS_STRIDE + bb]);
    }
    wait_async0();
    __syncthreads();

    const int b = tid & (B_ - 1);                       // batch row: 0..63
    const int g = tid >> 6;                             // table phase: 0..3
    // g is constant within every wave32 for this layout -> pin it to an SGPR
    // so mask/weight loads become scalar (constant-cache) loads.
    const int gu = __builtin_amdgcn_readfirstlane(g);

    const int2*   masks = (const int2*)input_mask;      // [T] index pairs
    const float4* w4    = (const float4*)lut_w;         // [T] corner weights

    for (int f = 0; f < OPB_; ++f) {
        const int  o     = blockIdx.x * OPB_ + f;
        const long tbase = (long)o * TPO_ + gu;

        const int2*   mp = masks + tbase;
        const float4* wp = w4    + tbase;

        float acc0 = 0.0f, acc1 = 0.0f;     // split chains for FMA ILP
        #pragma unroll 4
        for (int k = 0; k < TPO_ / 4; ++k) {
            const int2   m = mp[4 * k];                 // uniform -> s_load_b64
            const float4 w = wp[4 * k];                 // uniform -> s_load_b128

            // Wave-uniform coefficient transform (scalar-float eligible):
            const float u0 = w.x + w.y;     // (w0+w1)
            const float v0 = w.y - w.x;     // (w1-w0)
            const float u1 = w.z + w.w;     // (w2+w3)
            const float v1 = w.w - w.z;     // (w3-w2)

            const float x0 = xT[m.x * LDS_STRIDE + b];  // conflict-free ds_load
            const float x1 = xT[m.y * LDS_STRIDE + b];

            // per_table = (t0+t1) + x1*(t1-t0),  t0 = u0+x0*v0, t1 = u1+x0*v1
            const float t0 = fmaf(x0, v0, u0);
            const float t1 = fmaf(x0, v1, u1);
            acc0 += (t0 + t1);
            acc1  = fmaf(x1, (t1 - t0), acc1);
        }

        red[g * B_ + b] = acc0 + acc1;
        __syncthreads();
        if (tid < B_) {
            const float s = red[tid] + red[B_ + tid] +
                            red[2 * B_ + tid] + red[3 * B_ + tid];
            out[(long)tid * OUT_ + o] = 0.25f * s + bias[o];
        }
        __syncthreads();
    }
}

extern "C" void kernel_launch(void* const* d_in, const int* in_sizes, int n_in,
                              void* d_out, int out_size, void* d_ws, size_t ws_size,
                              hipStream_t stream) {
    const float* input      = (const float*)d_in[0];  // [64, 512]
    const int*   input_mask = (const int*)  d_in[1];  // [262144*2]
    const float* lut_w      = (const float*)d_in[2];  // [262144, 4]
    const float* bias       = (const float*)d_in[3];  // [512]
    float*       out        = (float*)d_out;          // [64, 512]

    dim3 grid(OUT_ / OPB_);      // 256 blocks -> fills a ~256-WGP-class chip
    dim3 block(NTHREADS);        // 256 threads = 8 wave32
    lut_linear_kernel<<<grid, block, 0, stream>>>(input, input_mask, lut_w, bias, out);
}